// MY_ADCRNN__30709016166898
// MI455X (gfx1250) — compile-verified
//
#include <hip/hip_runtime.h>
#include <hip/hip_bf16.h>

typedef __attribute__((ext_vector_type(2))) float v2f;
typedef __attribute__((ext_vector_type(8))) float v8f;

#define NNODES 100000
#define NEDGES 1000000
#define FDIM   32

// ---------------------------------------------------------------------------
// ws layout (floats):
//   [0,N)            deg_out
//   [N,2N)           deg_in
//   [2N,2N+E)        co   (norm_out per edge)
//   [2N+E,2N+2E)     ci   (norm_in  per edge)
//   then T1o, T1i, P2o, P2i  (N*32 each)
//   then Weff (10 * 32*32): g in {z,h}, j in 0..4
// ---------------------------------------------------------------------------

__global__ void k_zero(float* ws) {
    const long long degN  = 2LL * NNODES;
    const long long shift = 2LL * NEDGES;                 // skip co/ci (overwritten)
    const long long total = degN + 4LL * NNODES * FDIM;   // deg + T1o/T1i/P2o/P2i
    for (long long i = (long long)blockIdx.x * blockDim.x + threadIdx.x;
         i < total; i += (long long)gridDim.x * blockDim.x) {
        long long idx = (i < degN) ? i : i + shift;
        ws[idx] = 0.0f;
    }
}

__global__ void k_deg(const int* __restrict__ ei, const float* __restrict__ ew,
                      float* __restrict__ deg_out, float* __restrict__ deg_in) {
    int e = blockIdx.x * blockDim.x + threadIdx.x;
    if (e >= NEDGES) return;
    float w = ew[e];
    atomicAdd(&deg_out[ei[e]], w);           // src = ei[0][e]
    atomicAdd(&deg_in[ei[NEDGES + e]], w);   // dst = ei[1][e]
}

__global__ void k_coef(const int* __restrict__ ei, const float* __restrict__ ew,
                       const float* __restrict__ deg_out, const float* __restrict__ deg_in,
                       float* __restrict__ co, float* __restrict__ ci) {
    int e = blockIdx.x * blockDim.x + threadIdx.x;
    if (e >= NEDGES) return;
    float w = ew[e];
    co[e] = w / deg_out[ei[e]];
    ci[e] = w / deg_in[ei[NEDGES + e]];
}

// out[agg[e]] += coef[e] * vin[gat[e]]   (one thread = one edge x 4 features)
__global__ void k_prop(const int* __restrict__ agg, const int* __restrict__ gat,
                       const float* __restrict__ coef, const float* __restrict__ vin,
                       float* __restrict__ vout) {
    long long tid = (long long)blockIdx.x * blockDim.x + threadIdx.x;
    if (tid >= (long long)NEDGES * 8) return;
    int e = (int)(tid >> 3);
    int q = (int)(tid & 7) * 4;
    float c = coef[e];
    const float4 v = *reinterpret_cast<const float4*>(vin + (long long)gat[e] * FDIM + q);
    float* o = vout + (long long)agg[e] * FDIM + q;
    atomicAdd(o + 0, c * v.x);
    atomicAdd(o + 1, c * v.y);
    atomicAdd(o + 2, c * v.z);
    atomicAdd(o + 3, c * v.w);
}

// Effective weights: rows 0:32 of each 64x32 slice; j=0 folds W[0,0]+W[1,0].
// W shape (2,3,64,32) row-major: W[dir,k][r,c] at (dir*3+k)*2048 + r*32 + c.
__global__ void k_weff(const float* __restrict__ Wz, const float* __restrict__ Wh,
                       float* __restrict__ weff) {
    int t = blockIdx.x * blockDim.x + threadIdx.x;
    if (t >= 10240) return;
    int g   = t / 5120;
    int rem = t % 5120;
    int j   = rem / 1024;
    int rc  = rem % 1024;              // r*32 + c, r < 32
    const float* W = g ? Wh : Wz;
    float v;
    if (j == 0) {
        v = W[rc] + W[3 * 2048 + rc];  // (dir=0,k=0) + (dir=1,k=0)
    } else {
        const int slot[5] = {0, 1, 4, 2, 5};   // j -> dir*3+k
        v = W[slot[j] * 2048 + rc];
    }
    weff[t] = v;
}

__device__ __forceinline__ v8f wmma4(v2f a, v2f b, v8f c) {
    return __builtin_amdgcn_wmma_f32_16x16x4_f32(false, a, false, b,
                                                 (short)0, c, false, false);
}

// One wave per 16-node tile. Accumulate Z_pre and Hh_pre (2 col-tiles each)
// over 5 input matrices with f32 WMMA, then fused GRU + relu + linear head.
__global__ void __launch_bounds__(256) k_gemm(
    const float* __restrict__ X,   const float* __restrict__ T1o,
    const float* __restrict__ T1i, const float* __restrict__ P2o,
    const float* __restrict__ P2i, const float* __restrict__ weff,
    const float* __restrict__ bz,  const float* __restrict__ bh,
    const float* __restrict__ Wl,  const float* __restrict__ bl,
    float* __restrict__ y)
{
    int lane = threadIdx.x & 31;
    int wid  = threadIdx.x >> 5;
    int tile = blockIdx.x * (blockDim.x >> 5) + wid;
    const int numTiles = (NNODES + 15) / 16;
    if (tile >= numTiles) return;               // uniform per wave

    int hi = lane >> 4;                          // lane half (K striping / M+8)
    int ln = lane & 15;
    int m  = tile * 16 + ln;                     // A-row this lane loads
    if (m >= NNODES) m = NNODES - 1;             // safe clamp (N%16==0 anyway)

    v8f az0 = {}, az1 = {}, ah0 = {}, ah1 = {};
    const float* wz = weff;                      // gate z: 5 * 1024
    const float* wh = weff + 5 * 1024;           // gate h

#pragma unroll
    for (int k0 = 0; k0 < 32; k0 += 4) {
        int kk = k0 + 2 * hi;                    // this lane's K pair
        long long base = (long long)m * FDIM + kk;
        float x0 = X[base], x1 = X[base + 1];
        v2f af[5];
        af[0] = v2f{x0, x1};
        af[1] = v2f{T1o[base], T1o[base + 1]};
        af[2] = v2f{T1i[base], T1i[base + 1]};
        af[3] = v2f{2.0f * P2o[base] - x0, 2.0f * P2o[base + 1] - x1};  // T2o
        af[4] = v2f{2.0f * P2i[base] - x0, 2.0f * P2i[base + 1] - x1};  // T2i
#pragma unroll
        for (int j = 0; j < 5; ++j) {
            const float* bzp = wz + j * 1024 + kk * 32;   // B[k][n], n = ln(+16)
            const float* bhp = wh + j * 1024 + kk * 32;
            v2f bz0 = v2f{bzp[ln],      bzp[32 + ln]};
            v2f bz1 = v2f{bzp[16 + ln], bzp[48 + ln]};
            v2f bh0 = v2f{bhp[ln],      bhp[32 + ln]};
            v2f bh1 = v2f{bhp[16 + ln], bhp[48 + ln]};
            az0 = wmma4(af[j], bz0, az0);
            az1 = wmma4(af[j], bz1, az1);
            ah0 = wmma4(af[j], bh0, ah0);
            ah1 = wmma4(af[j], bh1, ah1);
        }
    }

    // Epilogue: Z = sigmoid(.+bz), Hn = (1-Z)*tanh(.+bh), y = relu(Hn)@Wl + bl
    float wl0 = Wl[ln],  wl1 = Wl[16 + ln];
    float bzb0 = bz[ln], bzb1 = bz[16 + ln];
    float bhb0 = bh[ln], bhb1 = bh[16 + ln];
    float blv = bl[0];
#pragma unroll
    for (int r = 0; r < 8; ++r) {
        int mr = tile * 16 + r + 8 * hi;         // C/D row for this VGPR slot
        float z0  = 1.0f / (1.0f + expf(-(az0[r] + bzb0)));
        float z1  = 1.0f / (1.0f + expf(-(az1[r] + bzb1)));
        float hn0 = (1.0f - z0) * tanhf(ah0[r] + bhb0);
        float hn1 = (1.0f - z1) * tanhf(ah1[r] + bhb1);
        float p = fmaxf(hn0, 0.0f) * wl0 + fmaxf(hn1, 0.0f) * wl1;
        p += __shfl_xor(p, 1, 32);               // reduce 16 lanes of this half
        p += __shfl_xor(p, 2, 32);
        p += __shfl_xor(p, 4, 32);
        p += __shfl_xor(p, 8, 32);
        if (ln == 0 && mr < NNODES) y[mr] = p + blv;
    }
}

extern "C" void kernel_launch(void* const* d_in, const int* in_sizes, int n_in,
                              void* d_out, int out_size, void* d_ws, size_t ws_size,
                              hipStream_t stream) {
    const float* x  = (const float*)d_in[0];
    const int*   ei = (const int*)  d_in[1];
    const float* ew = (const float*)d_in[2];
    // d_in[3]=h, d_in[4]=c : unused (reference hard-codes H0 = zeros)
    const float* Wz = (const float*)d_in[5];
    const float* bz = (const float*)d_in[6];
    // d_in[7]=Wr, d_in[8]=br : dead code when H0 == 0 (R*H0 == 0)
    const float* Wh = (const float*)d_in[9];
    const float* bh = (const float*)d_in[10];
    const float* Wl = (const float*)d_in[11];
    const float* bl = (const float*)d_in[12];
    float* y = (float*)d_out;

    float* ws      = (float*)d_ws;
    float* deg_out = ws;
    float* deg_in  = ws + NNODES;
    float* co      = ws + 2LL * NNODES;
    float* ci      = co + NEDGES;
    float* T1o     = ci + NEDGES;
    float* T1i     = T1o + (long long)NNODES * FDIM;
    float* P2o     = T1i + (long long)NNODES * FDIM;
    float* P2i     = P2o + (long long)NNODES * FDIM;
    float* weff    = P2i + (long long)NNODES * FDIM;

    const int* src = ei;            // edge_index[0]
    const int* dst = ei + NEDGES;   // edge_index[1]

    k_zero<<<2048, 256, 0, stream>>>(ws);

    int egrid = (NEDGES + 255) / 256;
    k_deg <<<egrid, 256, 0, stream>>>(ei, ew, deg_out, deg_in);
    k_coef<<<egrid, 256, 0, stream>>>(ei, ew, deg_out, deg_in, co, ci);

    long long pthreads = (long long)NEDGES * 8;
    int pgrid = (int)((pthreads + 255) / 256);
    k_prop<<<pgrid, 256, 0, stream>>>(dst, src, co, x,   T1o);  // prop_out(X)
    k_prop<<<pgrid, 256, 0, stream>>>(src, dst, ci, x,   T1i);  // prop_in(X)
    k_prop<<<pgrid, 256, 0, stream>>>(dst, src, co, T1o, P2o);  // prop_out(T1o)
    k_prop<<<pgrid, 256, 0, stream>>>(src, dst, ci, T1i, P2i);  // prop_in(T1i)

    k_weff<<<(10240 + 255) / 256, 256, 0, stream>>>(Wz, Wh, weff);

    int numTiles = (NNODES + 15) / 16;
    k_gemm<<<(numTiles + 7) / 8, 256, 0, stream>>>(x, T1o, T1i, P2o, P2i,
                                                   weff, bz, bh, Wl, bl, y);
}